// EProPnPBase_73443940762015
// MI455X (gfx1250) — compile-verified
//
#include <hip/hip_runtime.h>
#include <hip/hip_bf16.h>
#include <stdint.h>

// ---------------- problem constants (match reference) ----------------
#define BATCH       128
#define NPTS        512
#define MC_SAMPLES  512
#define NUM_ITER    4
#define ITER_S      128          // samples per iteration
#define EPSF        1e-5f
#define LOG2PI_F    1.8378770664093453f

// output layout (flat concatenation in reference return order)
#define OUT_POSE_OPT   0                          // (B,4)
#define OUT_COST       (OUT_POSE_OPT + BATCH*4)   // (B,)
#define OUT_POSE_PLUS  (OUT_COST + BATCH)         // (B,4)
#define OUT_SAMPLES    (OUT_POSE_PLUS + BATCH*4)  // (512,B,4)
#define OUT_LW         (OUT_SAMPLES + MC_SAMPLES*BATCH*4) // (512,B)
#define OUT_COST_INIT  (OUT_LW + MC_SAMPLES*BATCH)        // (B,)

#define SSTRIDE 5   // padded sample stride (odd -> LDS bank-conflict free)

typedef __attribute__((ext_vector_type(2))) float v2f;
typedef __attribute__((ext_vector_type(8))) float v8f;
typedef __attribute__((vector_size(4 * sizeof(int)))) int v4i_t;

#if __has_builtin(__builtin_amdgcn_wmma_f32_16x16x4_f32)
#define HAVE_WMMA_F32 1
#else
#define HAVE_WMMA_F32 0
#endif

#if __has_builtin(__builtin_amdgcn_global_load_async_to_lds_b128)
#define HAVE_ASYNC_LDS 1
#else
#define HAVE_ASYNC_LDS 0
#endif

// ---------------- small device helpers ----------------
__device__ __forceinline__ uint32_t pcg_hash(uint32_t x) {
    x = x * 747796405u + 2891336453u;
    uint32_t w = ((x >> ((x >> 28u) + 4u)) ^ x) * 277803737u;
    return (w >> 22u) ^ w;
}
__device__ __forceinline__ float u01(uint32_t h) {
    return (float)(h >> 8) * (1.0f / 16777216.0f) + (0.5f / 16777216.0f);
}

__device__ __forceinline__ void stage_to_lds(float* dst, const float* src,
                                             int n_f4, int tid) {
#if HAVE_ASYNC_LDS
    for (int i = tid; i < n_f4; i += 256) {
        __builtin_amdgcn_global_load_async_to_lds_b128(
            (v4i_t*)(src + i * 4),   // global source (const dropped; generic int4*)
            (v4i_t*)(dst + i * 4),   // LDS destination
            0, 0);
    }
#else
    const float4* s4 = reinterpret_cast<const float4*>(src);
    float4*       d4 = reinterpret_cast<float4*>(dst);
    for (int i = tid; i < n_f4; i += 256) d4[i] = s4[i];
#endif
}

__device__ __forceinline__ void wait_stage() {
#if HAVE_ASYNC_LDS
#if __has_builtin(__builtin_amdgcn_s_wait_asynccnt)
    __builtin_amdgcn_s_wait_asynccnt(0);
#else
    asm volatile("s_wait_asynccnt 0" ::: "memory");
#endif
#endif
}

__device__ __forceinline__ float block_sum(float v, volatile float* buf) {
    int t = threadIdx.x;
    buf[t] = v; __syncthreads();
    for (int o = 128; o > 0; o >>= 1) {
        if (t < o) buf[t] += buf[t + o];
        __syncthreads();
    }
    float r = buf[0]; __syncthreads();
    return r;
}
__device__ __forceinline__ float block_max(float v, volatile float* buf) {
    int t = threadIdx.x;
    buf[t] = v; __syncthreads();
    for (int o = 128; o > 0; o >>= 1) {
        if (t < o) buf[t] = fmaxf(buf[t], buf[t + o]);
        __syncthreads();
    }
    float r = buf[0]; __syncthreads();
    return r;
}

__device__ __forceinline__ float point_cost(const float* __restrict__ x3,
                                            const float* __restrict__ x2,
                                            const float* __restrict__ w2,
                                            int p, float cn, float sn,
                                            float tx, float ty, float tz,
                                            float fx, float fy, float cx, float cy) {
    float X = x3[p * 3 + 0], Y = x3[p * 3 + 1], Z = x3[p * 3 + 2];
    float Xc = fmaf(cn, X, fmaf(sn, Z, tx));
    float Yc = Y + ty;
    float Zc = fmaxf(fmaf(-sn, X, fmaf(cn, Z, tz)), EPSF);
    float inv = __builtin_amdgcn_rcpf(Zc);
    float u = fmaf(fx * Xc, inv, cx);
    float v = fmaf(fy * Yc, inv, cy);
    float ru = (u - x2[p * 2 + 0]) * w2[p * 2 + 0];
    float rv = (v - x2[p * 2 + 1]) * w2[p * 2 + 1];
    return fmaf(ru, ru, rv * rv);
}

// ---------------- fused AMIS kernel: one block per batch element ----------------
__global__ __launch_bounds__(256) void amis_kernel(
    const float* __restrict__ x3d, const float* __restrict__ x2d,
    const float* __restrict__ w2d, const float* __restrict__ cam,
    const float* __restrict__ pose_init, float* __restrict__ out) {

    __shared__ __align__(16) float s_x3d[NPTS * 3];
    __shared__ __align__(16) float s_x2d[NPTS * 2];
    __shared__ __align__(16) float s_w2d[NPTS * 2];
    __shared__ float s_samples[MC_SAMPLES * SSTRIDE];
    __shared__ float s_cost[MC_SAMPLES];
    __shared__ float s_lw[MC_SAMPLES];
    __shared__ float s_params[NUM_ITER * 8];   // per component: tm[3], ts[3], rm, rs
    __shared__ float s_red[256];

    const int b   = blockIdx.x;
    const int tid = threadIdx.x;

    stage_to_lds(s_x3d, x3d + (size_t)b * NPTS * 3, (NPTS * 3) / 4, tid);
    stage_to_lds(s_x2d, x2d + (size_t)b * NPTS * 2, (NPTS * 2) / 4, tid);
    stage_to_lds(s_w2d, w2d + (size_t)b * NPTS * 2, (NPTS * 2) / 4, tid);
    wait_stage();

    const float fx = cam[b * 4 + 0], fy = cam[b * 4 + 1];
    const float cx = cam[b * 4 + 2], cy = cam[b * 4 + 3];
    const float p0x = pose_init[b * 4 + 0], p0y = pose_init[b * 4 + 1];
    const float p0z = pose_init[b * 4 + 2], p0r = pose_init[b * 4 + 3];

    if (tid == 0) {
        s_params[0] = p0x; s_params[1] = p0y; s_params[2] = p0z;
        s_params[3] = 0.1f; s_params[4] = 0.1f; s_params[5] = 0.1f;
        s_params[6] = p0r; s_params[7] = 0.2f;
    }
    __syncthreads();   // also covers LDS staging visibility across waves

    // ---- cost_init = cost(pose_init) ----
    float sn0, cn0; __sincosf(p0r, &sn0, &cn0);
    float acc = 0.0f;
    for (int p = tid; p < NPTS; p += 256)
        acc += point_cost(s_x3d, s_x2d, s_w2d, p, cn0, sn0, p0x, p0y, p0z, fx, fy, cx, cy);
    const float cost_init = 0.5f * block_sum(acc, s_red);

    // ---- AMIS iterations ----
    for (int it = 0; it < NUM_ITER; ++it) {
        // 1) draw ITER_S samples from params[it]
        if (tid < ITER_S) {
            const float* pp = &s_params[it * 8];
            const int sidx = it * ITER_S + tid;
            float z[4];
            #pragma unroll
            for (int j = 0; j < 2; ++j) {
                uint32_t h = ((uint32_t)b * 0x9E3779B9u) ^ ((uint32_t)sidx * 0x85EBCA77u)
                           ^ ((uint32_t)j * 0xC2B2AE3Du) ^ 0x01234567u;
                uint32_t r1 = pcg_hash(h);
                uint32_t r2 = pcg_hash(h ^ 0x68BC21EBu);
                float u1 = fmaxf(u01(r1), 1e-9f);
                float u2 = u01(r2);
                float rad = __fsqrt_rn(-2.0f * __logf(u1));
                float a = 6.283185307179586f * u2;
                float sa, ca; __sincosf(a, &sa, &ca);
                z[2 * j + 0] = rad * ca;
                z[2 * j + 1] = rad * sa;
            }
            s_samples[sidx * SSTRIDE + 0] = fmaf(pp[3], z[0], pp[0]);
            s_samples[sidx * SSTRIDE + 1] = fmaf(pp[4], z[1], pp[1]);
            s_samples[sidx * SSTRIDE + 2] = fmaf(pp[5], z[2], pp[2]);
            s_samples[sidx * SSTRIDE + 3] = fmaf(pp[7], z[3], pp[6]);
        }
        __syncthreads();

        // 2) cost of each sample: 16 samples per wave, 2 lanes/sample, 2 accs/lane.
        {
            const int lane = tid & 31;
            const int wv   = tid >> 5;
            const int m    = lane & 15;
            const int half = lane >> 4;
            const int sidx = it * ITER_S + wv * 16 + m;
            const float tx  = s_samples[sidx * SSTRIDE + 0];
            const float ty  = s_samples[sidx * SSTRIDE + 1];
            const float tz  = s_samples[sidx * SSTRIDE + 2];
            const float yaw = s_samples[sidx * SSTRIDE + 3];
            float sn, cn; __sincosf(yaw, &sn, &cn);
            float a0 = 0.0f, a1 = 0.0f;
            const int pb = half * 256;
            #pragma unroll 4
            for (int p = 0; p < 128; ++p)
                a0 += point_cost(s_x3d, s_x2d, s_w2d, pb + p, cn, sn, tx, ty, tz, fx, fy, cx, cy);
            #pragma unroll 4
            for (int p = 128; p < 256; ++p)
                a1 += point_cost(s_x3d, s_x2d, s_w2d, pb + p, cn, sn, tx, ty, tz, fx, fy, cx, cy);

#if HAVE_WMMA_F32
            // A(16x4) = per-sample partial sums (rows = samples, K = lane-half x acc).
            // B = ones(4x16)  ->  D[m][n] = sum_k A[m][k] = total cost of sample m.
            v2f A; A.x = a0; A.y = a1;
            v2f Bv; Bv.x = 1.0f; Bv.y = 1.0f;
            v8f Cv = {};
            Cv = __builtin_amdgcn_wmma_f32_16x16x4_f32(
                     false, A, false, Bv, (short)0, Cv, false, false);
            if (m == 0) {   // lane 0 holds rows 0..7, lane 16 holds rows 8..15
                const int base = it * ITER_S + wv * 16 + half * 8;
                #pragma unroll
                for (int r = 0; r < 8; ++r)
                    s_cost[base + r] = 0.5f * Cv[r];
            }
#else
            float t = a0 + a1;
            t += __shfl_xor(t, 16);
            if (half == 0)
                s_cost[it * ITER_S + wv * 16 + m] = 0.5f * t;
#endif
        }
        __syncthreads();

        // 3) mixture logprobs + logweights over ALL accumulated samples
        const int S = (it + 1) * ITER_S;
        for (int s = tid; s < S; s += 256) {
            const float x0 = s_samples[s * SSTRIDE + 0];
            const float x1 = s_samples[s * SSTRIDE + 1];
            const float x2 = s_samples[s * SSTRIDE + 2];
            const float x3 = s_samples[s * SSTRIDE + 3];
            float lps[NUM_ITER];
            float mx = -3.4e38f;
            for (int k = 0; k <= it; ++k) {
                const float* pk = &s_params[k * 8];
                float d0 = (x0 - pk[0]) / pk[3];
                float d1 = (x1 - pk[1]) / pk[4];
                float d2 = (x2 - pk[2]) / pk[5];
                float tlp = -0.5f * (d0 * d0 + d1 * d1 + d2 * d2)
                          - (__logf(pk[3]) + __logf(pk[4]) + __logf(pk[5]))
                          - 1.5f * LOG2PI_F;
                float dr = (x3 - pk[6]) / pk[7];
                float rlp = -0.5f * dr * dr - __logf(pk[7]) - 0.5f * LOG2PI_F;
                lps[k] = tlp + rlp;
                mx = fmaxf(mx, lps[k]);
            }
            float se = 0.0f;
            for (int k = 0; k <= it; ++k) se += __expf(lps[k] - mx);
            float mix = mx + __logf(se) - __logf((float)(it + 1));
            s_lw[s] = -s_cost[s] - mix;
        }
        __syncthreads();

        // 4) softmax-weighted moment update for next proposal
        if (it < NUM_ITER - 1) {
            float lm = -3.4e38f;
            for (int s = tid; s < S; s += 256) lm = fmaxf(lm, s_lw[s]);
            const float gmax = block_max(lm, s_red);

            float pw = 0, px0 = 0, px1 = 0, px2 = 0, px3 = 0;
            for (int s = tid; s < S; s += 256) {
                float e = __expf(s_lw[s] - gmax);
                pw  += e;
                px0 += e * s_samples[s * SSTRIDE + 0];
                px1 += e * s_samples[s * SSTRIDE + 1];
                px2 += e * s_samples[s * SSTRIDE + 2];
                px3 += e * s_samples[s * SSTRIDE + 3];
            }
            const float sw  = block_sum(pw,  s_red);
            const float sx0 = block_sum(px0, s_red);
            const float sx1 = block_sum(px1, s_red);
            const float sx2 = block_sum(px2, s_red);
            const float sx3 = block_sum(px3, s_red);
            const float inv = 1.0f / sw;
            const float m0 = sx0 * inv, m1 = sx1 * inv, m2 = sx2 * inv, m3 = sx3 * inv;

            float v0 = 0, v1 = 0, v2 = 0, v3 = 0;
            for (int s = tid; s < S; s += 256) {
                float e = __expf(s_lw[s] - gmax);
                float d0 = s_samples[s * SSTRIDE + 0] - m0;
                float d1 = s_samples[s * SSTRIDE + 1] - m1;
                float d2 = s_samples[s * SSTRIDE + 2] - m2;
                float d3 = s_samples[s * SSTRIDE + 3] - m3;
                v0 += e * d0 * d0; v1 += e * d1 * d1;
                v2 += e * d2 * d2; v3 += e * d3 * d3;
            }
            const float gv0 = block_sum(v0, s_red);
            const float gv1 = block_sum(v1, s_red);
            const float gv2 = block_sum(v2, s_red);
            const float gv3 = block_sum(v3, s_red);

            if (tid == 0) {
                float* pn = &s_params[(it + 1) * 8];
                pn[0] = m0; pn[1] = m1; pn[2] = m2;
                pn[3] = __fsqrt_rn(gv0 * inv + EPSF);
                pn[4] = __fsqrt_rn(gv1 * inv + EPSF);
                pn[5] = __fsqrt_rn(gv2 * inv + EPSF);
                pn[6] = m3;
                pn[7] = __fsqrt_rn(gv3 * inv + EPSF);
            }
            __syncthreads();
        }
    }

    // ---- write outputs ----
    for (int idx = tid; idx < MC_SAMPLES * 4; idx += 256) {
        int s = idx >> 2, k = idx & 3;
        out[OUT_SAMPLES + ((size_t)s * BATCH + b) * 4 + k] = s_samples[s * SSTRIDE + k];
    }
    for (int s = tid; s < MC_SAMPLES; s += 256)
        out[OUT_LW + (size_t)s * BATCH + b] = s_lw[s];
    if (tid < 4) {
        float pv = pose_init[b * 4 + tid];
        out[OUT_POSE_OPT  + b * 4 + tid] = pv;
        out[OUT_POSE_PLUS + b * 4 + tid] = pv;
    }
    if (tid == 0) {
        out[OUT_COST      + b] = cost_init;
        out[OUT_COST_INIT + b] = cost_init;
    }
}

extern "C" void kernel_launch(void* const* d_in, const int* in_sizes, int n_in,
                              void* d_out, int out_size, void* d_ws, size_t ws_size,
                              hipStream_t stream) {
    (void)in_sizes; (void)n_in; (void)out_size; (void)d_ws; (void)ws_size;
    const float* x3d       = (const float*)d_in[0];
    const float* x2d       = (const float*)d_in[1];
    const float* w2d       = (const float*)d_in[2];
    const float* cam_intr  = (const float*)d_in[3];
    const float* pose_init = (const float*)d_in[4];
    float* out = (float*)d_out;

    amis_kernel<<<BATCH, 256, 0, stream>>>(x3d, x2d, w2d, cam_intr, pose_init, out);
}